// StudentTeacherLoss_80487687127344
// MI455X (gfx1250) — compile-verified
//
#include <hip/hip_runtime.h>
#include <hip/hip_bf16.h>

typedef float v2f __attribute__((ext_vector_type(2)));
typedef float v4f __attribute__((ext_vector_type(4)));
typedef float v8f __attribute__((ext_vector_type(8)));

// ---------------------------------------------------------------------------
// Full 32-lane wave sum using two chained V_WMMA_F32_16X16X4_F32 with a
// ones B-matrix. A-layout (16x4 f32): lane L<16 holds A[L][0..1] in its two
// A VGPRs, lane L>=16 holds A[L-16][2..3]. With A.x = p, A.y = 0:
//   D[m][*] = p[m] + p[m+16]           (first WMMA: pairwise half-wave fold)
// Summing the 8 C/D VGPRs yields a value uniform within each half-wave
// (lanes 0-15 hold sum of M=0..7 rows, lanes 16-31 hold M=8..15 rows).
// Feeding that back through the same trick gives lowHalf+highHalf = total,
// broadcast to every lane. No DS/permlane traffic, no divergence.
// ---------------------------------------------------------------------------
__device__ inline float wave_sum_wmma(float p) {
    v2f ones; ones.x = 1.0f; ones.y = 1.0f;
    v2f a;    a.x = p;       a.y = 0.0f;
    v8f c = {};
    c = __builtin_amdgcn_wmma_f32_16x16x4_f32(false, a, false, ones,
                                              (short)0, c, false, false);
    float s = c[0] + c[1] + c[2] + c[3] + c[4] + c[5] + c[6] + c[7];
    v2f a2; a2.x = s; a2.y = 0.0f;
    v8f c2 = {};
    c2 = __builtin_amdgcn_wmma_f32_16x16x4_f32(false, a2, false, ones,
                                               (short)0, c2, false, false);
    return c2[0];   // full-wave total, uniform across all 32 lanes
}

// ---------------------------------------------------------------------------
// One wave per ragged row. Streams the two ragged rows (visual + text) with
// nontemporal b128 loads (read-once -> NT hint keeps L2 free for the reused
// global rows), accumulates fused vis+txt squared error per lane, WMMA
// wave-reduce, then lane 0 pushes into per-segment accumulators.
// ---------------------------------------------------------------------------
__global__ void distill_rows_kernel(const float* __restrict__ rows_a,  // ragged visual rows
                                    const float* __restrict__ glob_a,  // global visual  [B,768]
                                    const float* __restrict__ rows_b,  // ragged text rows
                                    const float* __restrict__ glob_b,  // global text    [B,768]
                                    const int*   __restrict__ seg_ids,
                                    float* __restrict__ seg_sum,
                                    float* __restrict__ seg_cnt,
                                    int nrows)
{
    const int lane   = threadIdx.x & 31;
    const int wave   = blockIdx.x * (blockDim.x >> 5) + (threadIdx.x >> 5);
    const int nwaves = gridDim.x * (blockDim.x >> 5);
    const float invD = 1.0f / 768.0f;

    for (int row = wave; row < nrows; row += nwaves) {
        const int seg = seg_ids[row];
        const v4f* __restrict__ ra = (const v4f*)(rows_a + (size_t)row * 768);
        const v4f* __restrict__ rb = (const v4f*)(rows_b + (size_t)row * 768);
        const v4f* __restrict__ ga = (const v4f*)(glob_a + (size_t)seg * 768);
        const v4f* __restrict__ gb = (const v4f*)(glob_b + (size_t)seg * 768);

        float p = 0.0f;
        #pragma unroll
        for (int j = 0; j < 6; ++j) {           // 768 f32 = 192 float4 = 6 * 32 lanes
            const int idx = lane + 32 * j;
            v4f va  = __builtin_nontemporal_load(&ra[idx]);   // streamed, NT
            v4f vb  = __builtin_nontemporal_load(&rb[idx]);   // streamed, NT
            v4f vga = ga[idx];                                // reused 32x, cached
            v4f vgb = gb[idx];
            v4f d1 = va - vga;
            v4f d2 = vb - vgb;
            p += d1.x*d1.x + d1.y*d1.y + d1.z*d1.z + d1.w*d1.w
               + d2.x*d2.x + d2.y*d2.y + d2.z*d2.z + d2.w*d2.w;
        }

        const float tot = wave_sum_wmma(p);     // EXEC is all-1s here (wave-uniform loop)
        if (lane == 0) {
            atomicAdd(&seg_sum[seg], tot * invD);
            atomicAdd(&seg_cnt[seg], 1.0f);
        }
    }
}

__global__ void zero_ws_kernel(float* __restrict__ ws, int n) {
    const int i = blockIdx.x * blockDim.x + threadIdx.x;
    if (i < n) ws[i] = 0.0f;
}

// mean over segments of per-segment mean, obj + rel, single block of 1024
__global__ void finalize_kernel(const float* __restrict__ obj_sum,
                                const float* __restrict__ obj_cnt,
                                const float* __restrict__ rel_sum,
                                const float* __restrict__ rel_cnt,
                                float* __restrict__ out, int nseg)
{
    __shared__ float red[1024];
    const int t = threadIdx.x;
    float v = 0.0f;
    if (t < nseg) {
        v = obj_sum[t] / fmaxf(obj_cnt[t], 1.0f)
          + rel_sum[t] / fmaxf(rel_cnt[t], 1.0f);
    }
    red[t] = v;
    __syncthreads();
    #pragma unroll
    for (int s = 512; s > 0; s >>= 1) {
        if (t < s) red[t] += red[t + s];
        __syncthreads();
    }
    if (t == 0) out[0] = red[0] / (float)nseg;
}

extern "C" void kernel_launch(void* const* d_in, const int* in_sizes, int n_in,
                              void* d_out, int out_size, void* d_ws, size_t ws_size,
                              hipStream_t stream) {
    const float* gve = (const float*)d_in[0];   // [B,768] global visual
    const float* gte = (const float*)d_in[1];   // [B,768] global text
    const float* ove = (const float*)d_in[2];   // [N_OBJ,768]
    const float* ote = (const float*)d_in[3];   // [N_OBJ,768]
    const float* rve = (const float*)d_in[4];   // [N_REL,768]
    const float* rte = (const float*)d_in[5];   // [N_REL,768]
    const int*   oseg = (const int*)d_in[6];
    const int*   rseg = (const int*)d_in[7];

    const int nObj = in_sizes[6];
    const int nRel = in_sizes[7];
    const int nSeg = in_sizes[0] / 768;         // B = 1024

    float* ws       = (float*)d_ws;
    float* obj_sum  = ws;
    float* obj_cnt  = ws + nSeg;
    float* rel_sum  = ws + 2 * nSeg;
    float* rel_cnt  = ws + 3 * nSeg;

    // deterministic: re-zero accumulators every call
    const int nzero = 4 * nSeg;
    zero_ws_kernel<<<(nzero + 255) / 256, 256, 0, stream>>>(ws, nzero);

    // one row per wave; 256 threads = 8 waves/block
    const int waves_needed = nObj;              // 32768
    const int blocks = (waves_needed + 7) / 8;  // 4096 blocks -> single pass

    distill_rows_kernel<<<blocks, 256, 0, stream>>>(
        ove, gve, ote, gte, oseg, obj_sum, obj_cnt, nObj);
    distill_rows_kernel<<<blocks, 256, 0, stream>>>(
        rve, gve, rte, gte, rseg, rel_sum, rel_cnt, nRel);

    finalize_kernel<<<1, 1024, 0, stream>>>(
        obj_sum, obj_cnt, rel_sum, rel_cnt, (float*)d_out, nSeg);
}